// CovarianceWeightedMSELoss_55087250539214
// MI455X (gfx1250) — compile-verified
//
#include <hip/hip_runtime.h>

typedef float v2f __attribute__((ext_vector_type(2)));
typedef float v4f __attribute__((ext_vector_type(4)));
typedef float v8f __attribute__((ext_vector_type(8)));

#define D_DIM 16
#define HW    55296                 // H*W = 192*288
#define N_TOT (32 * HW)             // 1,769,472
#define NT    256                   // n-columns per tile
#define TILES_PER_BATCH (HW / NT)   // 216 (exact)
#define NTILES (N_TOT / NT)         // 6912
#define LDS_STRIDE 260              // 256 + 4 pad -> conflict-free float2 reads
#define TPB   256                   // 8 waves (wave32)
#define WS_PER_BLOCK 272            // 256 S-partials + 16 mu-partials

// ---------------------------------------------------------------------------
// Stage 1: stream residuals, accumulate S = R*R^T via v_wmma_f32_16x16x4_f32
// and mu = row sums. One partial (S[256], mu[16]) per block into d_ws.
// ---------------------------------------------------------------------------
__global__ __launch_bounds__(TPB)
void cov_stage1(const float* __restrict__ yt, const float* __restrict__ yp,
                float* __restrict__ ws)
{
    __shared__ __align__(16) float lds[D_DIM * LDS_STRIDE];  // 16640 B

    const int t    = threadIdx.x;
    const int wave = t >> 5;
    const int lane = t & 31;
    const int arow = lane & 15;          // matrix row this lane feeds (A and B)
    const int kofs = (lane >> 4) << 1;   // lanes 16-31 hold K=2,3

    v8f   acc = {0.f, 0.f, 0.f, 0.f, 0.f, 0.f, 0.f, 0.f};
    float mu  = 0.f;

    for (int tt = blockIdx.x; tt < NTILES; tt += gridDim.x) {
        // tile tt covers n in [tt*256, tt*256+256): never crosses a batch
        const int    b    = tt / TILES_PER_BATCH;
        const int    hw0  = (tt - b * TILES_PER_BATCH) * NT;
        const size_t base = (size_t)b * (D_DIM * HW) + (size_t)hw0;

        __syncthreads();  // previous tile's WMMA reads done before overwrite
        #pragma unroll
        for (int rep = 0; rep < 4; ++rep) {
            const int    u   = rep * TPB + t;
            const int    row = u >> 6;          // 0..15
            const int    q   = (u & 63) << 2;   // 0..252, step 4
            const size_t g   = base + (size_t)row * HW + (size_t)q;
            v4f a = *(const v4f*)(yt + g);      // coalesced global_load_b128
            v4f p = *(const v4f*)(yp + g);
            v4f r = a - p;
            *(v4f*)&lds[row * LDS_STRIDE + q] = r;
        }
        __syncthreads();

        // each wave owns a 32-column slice; 8 WMMA steps of K=4
        const int cbase = wave * 32;
        #pragma unroll
        for (int kk = 0; kk < 8; ++kk) {
            const int col = cbase + (kk << 2) + kofs;
            v2f v = *(const v2f*)&lds[arow * LDS_STRIDE + col];
            mu += v.x + v.y;  // each column counted exactly once per wave
            // A (16x4) and B (4x16 = A^T) have identical per-lane contents
            acc = __builtin_amdgcn_wmma_f32_16x16x4_f32(
                false, v, false, v, (short)0, acc, false, false);
        }
    }

    // -------- block-level reduction of the 8 per-wave accumulators --------
    __syncthreads();
    #pragma unroll
    for (int r = 0; r < 8; ++r) {
        // C/D layout: VGPR r, lanes 0-15 -> M=r, lanes 16-31 -> M=r+8; N=lane%16
        const int m = r + ((lane >> 4) << 3);
        lds[wave * 256 + m * 16 + (lane & 15)] = acc[r];
    }
    mu += __shfl_xor(mu, 16, 32);            // fold K-halves: lanes 0-15 own d
    if (lane < 16) lds[2048 + wave * 16 + lane] = mu;
    __syncthreads();

    float s = 0.f;
    #pragma unroll
    for (int w = 0; w < 8; ++w) s += lds[w * 256 + t];
    ws[(size_t)blockIdx.x * WS_PER_BLOCK + t] = s;
    if (t < 16) {
        float m = 0.f;
        #pragma unroll
        for (int w = 0; w < 8; ++w) m += lds[2048 + w * 16 + t];
        ws[(size_t)blockIdx.x * WS_PER_BLOCK + 256 + t] = m;
    }
}

// ---------------------------------------------------------------------------
// Stage 2: sum partials, cov = (S - mu mu^T/N)/(N-1), P = cov^-1 (Gauss-
// Jordan, SPD so no pivoting), out = trace(P*S)/N.
// ---------------------------------------------------------------------------
__global__ __launch_bounds__(256)
void cov_stage2(const float* __restrict__ ws, int nblocks,
                float* __restrict__ out)
{
    __shared__ float S[256];
    __shared__ float A[256];
    __shared__ float P[256];
    __shared__ float mu[16];
    __shared__ float red[256];

    const int t = threadIdx.x;
    const int d = t >> 4;
    const int e = t & 15;

    float s = 0.f;
    for (int b = 0; b < nblocks; ++b) s += ws[(size_t)b * WS_PER_BLOCK + t];
    S[t] = s;
    if (t < 16) {
        float m = 0.f;
        for (int b = 0; b < nblocks; ++b) m += ws[(size_t)b * WS_PER_BLOCK + 256 + t];
        mu[t] = m;
    }
    __syncthreads();

    const float Nf = (float)N_TOT;
    A[t] = (S[t] - mu[d] * mu[e] / Nf) / (Nf - 1.0f);
    P[t] = (d == e) ? 1.f : 0.f;
    __syncthreads();

    for (int k = 0; k < 16; ++k) {
        const float piv = A[k * 16 + k];
        __syncthreads();
        if (d == k) { A[t] /= piv; P[t] /= piv; }
        __syncthreads();
        const float f   = A[d * 16 + k];
        const float ake = A[k * 16 + e];
        const float pke = P[k * 16 + e];
        __syncthreads();
        if (d != k) { A[t] -= f * ake; P[t] -= f * pke; }
        __syncthreads();
    }

    red[t] = P[t] * S[e * 16 + d];   // trace(P * S)
    __syncthreads();
    for (int off = 128; off > 0; off >>= 1) {
        if (t < off) red[t] += red[t + off];
        __syncthreads();
    }
    if (t == 0) out[0] = red[0] / Nf;
}

// ---------------------------------------------------------------------------
extern "C" void kernel_launch(void* const* d_in, const int* in_sizes, int n_in,
                              void* d_out, int out_size, void* d_ws, size_t ws_size,
                              hipStream_t stream)
{
    (void)in_sizes; (void)n_in; (void)out_size;
    const float* yt  = (const float*)d_in[0];
    const float* yp  = (const float*)d_in[1];
    float*       out = (float*)d_out;
    float*       ws  = (float*)d_ws;

    // one partial record (272 floats) per block; cap grid by scratch capacity
    long long cap = (long long)(ws_size / (WS_PER_BLOCK * sizeof(float)));
    int nb = (int)(cap < 1024 ? cap : 1024);
    if (nb < 1) nb = 1;
    if (nb > NTILES) nb = NTILES;

    cov_stage1<<<nb, TPB, 0, stream>>>(yt, yp, ws);
    cov_stage2<<<1, 256, 0, stream>>>(ws, nb, out);
}